// GaussianSmoothing_86199993631296
// MI455X (gfx1250) — compile-verified
//
#include <hip/hip_runtime.h>

typedef __attribute__((ext_vector_type(2))) float v2f;
typedef __attribute__((ext_vector_type(4))) float v4f;
typedef __attribute__((ext_vector_type(8))) float v8f;

#define HM 64
#define KP 24
#define HH (HM * HM)

// One workgroup per batch image. 256 threads = 8 wave32.
// Phase 1: build the summed 64x64 delta tile in LDS (splat 5x5 gaussians).
// Phase 2: broadcast tile to all 24 output channels via v_wmma_f32_16x16x4_f32
//          (D[m][n] = 1.0*pixel[m], exact) + nontemporal b128 stores.
__global__ __launch_bounds__(256) void GaussianSmoothing_kernel(
    const float* __restrict__ x,          // [B,24,2]
    const float* __restrict__ weight,     // [24,24,5,5] (all slices identical)
    const int*   __restrict__ vis_batch,  // [256]
    const int*   __restrict__ vis_kps,    // [256]
    float*       __restrict__ out)        // [B,24,64,64]
{
    __shared__ float    S[HH];
    __shared__ float    gl[25];
    __shared__ int      pos[KP];          // cy*64+cx, or -1 if suppressed
    __shared__ unsigned killmask;

    const int t = threadIdx.x;
    const int b = blockIdx.x;

    // --- init LDS ---
    #pragma unroll
    for (int i = 0; i < HH / 256; ++i) S[t + i * 256] = 0.0f;
    if (t < 25) gl[t] = weight[t];          // weight[0][0][:,:] slice
    if (t == 0) killmask = 0u;
    __syncthreads();

    // --- kill mask from the 256 (vis_batch, vis_kps) pairs ---
    {
        int vb = vis_batch[t];
        if (vb == b) atomicOr(&killmask, 1u << (vis_kps[t] & 31));
    }
    __syncthreads();

    // --- per-keypoint coords (computed once) ---
    if (t < KP) {
        float xx = x[(b * KP + t) * 2 + 0];
        float yy = x[(b * KP + t) * 2 + 1];
        // jnp.round == round-half-to-even == rintf (default RNE mode)
        int cx = (int)rintf(((xx + 1.0f) * 0.5f) * 63.0f);
        int cy = (int)rintf(((yy + 1.0f) * 0.5f) * 63.0f);
        bool valid = (cx >= 0) && (cx < HM) && (cy >= 0) && (cy < HM);
        // invalid rows -> coords (0,0) -> cx==0 -> never placed anyway
        bool place = valid && (cx != 0) && !((killmask >> t) & 1u);
        pos[t] = place ? (cy * HM + cx) : -1;
    }
    __syncthreads();

    // --- splat: 24 keypoints x 25 taps = 600 tasks ---
    for (int task = t; task < KP * 25; task += 256) {
        int k   = task / 25;
        int tap = task - k * 25;
        int p   = pos[k];
        if (p >= 0) {
            int cy = p >> 6, cx = p & 63;
            int dy = tap / 5, dx = tap - dy * 5;
            int ty = cy + dy - 2, tx = cx + dx - 2;   // g is 180-deg symmetric
            if (ty >= 0 && ty < HM && tx >= 0 && tx < HM)
                atomicAdd(&S[ty * HM + tx], gl[tap]);
        }
    }
    __syncthreads();

    // --- broadcast tile to 24 channels via WMMA (exact rank-1 matmul) ---
    const int  lane    = t & 31;
    const int  wave    = t >> 5;
    const bool lo_half = lane < 16;
    float* outb = out + (size_t)b * KP * HH;

    // contiguous 32-group block per wave -> each wave streams a sequential
    // 2KB run per channel (better GL2 write combining for the NT stream)
    const int g0 = wave * (HH / 16 / 8);
    for (int grp = g0; grp < g0 + (HH / 16 / 8); ++grp) {
        const int p0 = grp * 16;
        // A[16x4]: column 0 = 16 pixel values, rest 0.
        // Layout: VGPR0 lanes0-15 K=0 (M=lane), lanes16-31 K=2; VGPR1 K=1/K=3.
        v2f a  = { lo_half ? S[p0 + lane] : 0.0f, 0.0f };
        // B[4x16]: row 0 = ones (VGPR0 lanes0-15), rest 0.
        v2f bm = { lo_half ? 1.0f : 0.0f, 0.0f };
        v8f c  = {};
        v8f d  = __builtin_amdgcn_wmma_f32_16x16x4_f32(
                     false, a, false, bm, (short)0, c, false, false);
        // D[m][n] = S[p0+m] for every column n.
        // Lane n<16:  d[r] = S[p0+r]   , "channel" = n
        // Lane n>=16: d[r] = S[p0+8+r] , "channel" = n-16
        v4f dlo = { d[0], d[1], d[2], d[3] };
        v4f dhi = { d[4], d[5], d[6], d[7] };
        const int ch   = lane & 15;
        const int poff = lo_half ? 0 : 8;
        float* p = outb + ch * HH + p0 + poff;
        __builtin_nontemporal_store(dlo, (v4f*)p);
        __builtin_nontemporal_store(dhi, (v4f*)(p + 4));
        if (ch < 8) {   // channels 16..23 reuse identical register data
            float* q = outb + (16 + ch) * HH + p0 + poff;
            __builtin_nontemporal_store(dlo, (v4f*)q);
            __builtin_nontemporal_store(dhi, (v4f*)(q + 4));
        }
    }
}

extern "C" void kernel_launch(void* const* d_in, const int* in_sizes, int n_in,
                              void* d_out, int out_size, void* d_ws, size_t ws_size,
                              hipStream_t stream) {
    const float* x         = (const float*)d_in[0];
    const float* weight    = (const float*)d_in[1];
    const int*   vis_batch = (const int*)d_in[2];
    const int*   vis_kps   = (const int*)d_in[3];
    float*       out       = (float*)d_out;

    const int B = in_sizes[0] / (KP * 2);   // 1024
    GaussianSmoothing_kernel<<<dim3(B), dim3(256), 0, stream>>>(
        x, weight, vis_batch, vis_kps, out);
}